// ProductVectorQuantizer_83923660964605
// MI455X (gfx1250) — compile-verified
//
#include <hip/hip_runtime.h>

#define N_ROWS   65536
#define E_DIM_   256
#define NQ_      8
#define D_SUB_   32
#define N_E_     512
#define ROWS_PER_BLOCK 128   // 8 waves * 16 rows

typedef float v2f __attribute__((ext_vector_type(2)));
typedef float v8f __attribute__((ext_vector_type(8)));

// LDS codebook layout: entry e, dim d stored at  e*32 + swz(e, d)
// pair-granular XOR swizzle keeps b64 alignment and removes bank conflicts
// for the A-operand fetch pattern (16 consecutive entries, fixed dim pair).
__device__ __forceinline__ int swz(int e, int d) {
    return ((((d >> 1) ^ (e & 15)) << 1) | (d & 1));
}

// ---------------------------------------------------------------------------
// Pre-kernel: codebook squared norms -> d_ws[0 .. NQ*N_E)
// ---------------------------------------------------------------------------
__global__ __launch_bounds__(256) void pvq_norms(const float* __restrict__ cb,
                                                 float* __restrict__ norms) {
    int e = blockIdx.x * 256 + threadIdx.x;           // 0 .. NQ*N_E-1
    if (e < NQ_ * N_E_) {
        const float* c = cb + (size_t)e * D_SUB_;
        float s = 0.f;
        #pragma unroll
        for (int d = 0; d < D_SUB_; ++d) s += c[d] * c[d];
        norms[e] = s;
    }
}

// ---------------------------------------------------------------------------
// Main kernel: WMMA f32 16x16x4 distance scores + argmin + gather + loss part
// ---------------------------------------------------------------------------
__global__ __launch_bounds__(256) void pvq_main(const float* __restrict__ z,
                                                const float* __restrict__ cb,
                                                const float* __restrict__ norms,
                                                float* __restrict__ out,
                                                float* __restrict__ partial) {
    __shared__ float lds_cb[N_E_ * D_SUB_];   // exactly 64 KB: -2 * codebook[q]

    const int q    = blockIdx.y;
    const int tid  = threadIdx.x;
    const int lane = tid & 31;
    const int wave = tid >> 5;

    const float* cbq = cb + (size_t)q * (N_E_ * D_SUB_);

    // Stage -2*codebook into LDS (swizzled). Coalesced global reads.
    for (int i = tid; i < N_E_ * D_SUB_; i += 256) {
        int e = i >> 5, d = i & 31;
        lds_cb[(e << 5) + swz(e, d)] = -2.0f * cbq[i];
    }
    __syncthreads();

    const int rowBase = blockIdx.x * ROWS_PER_BLOCK + wave * 16;
    const int nrow    = lane & 15;   // N index (row within tile)
    const int khalf   = lane >> 4;   // half-wave selects K (A/B) or M+8 (C/D)

    // Preload B operand: z^T tile, loop-invariant. 8 chunks of K=4.
    // B[4x16] lane layout: n = lane%16, k = chunk*4 + (lane/16)*2 + {0,1}
    v2f breg[8];
    const float* zrow = z + (size_t)(rowBase + nrow) * E_DIM_ + q * D_SUB_;
    #pragma unroll
    for (int kk = 0; kk < 8; ++kk)
        breg[kk] = *(const v2f*)(zrow + kk * 4 + khalf * 2);

    const float* nrm_q = norms + q * N_E_;

    float minS = 3.0e38f;
    int   minI = 0;

    for (int kt = 0; kt < N_E_ / 16; ++kt) {
        const int mbase = kt * 16 + khalf * 8;   // C/D: lane holds M = mbase+v

        // C preloaded with ||c||^2  ->  D = ||c||^2 - 2 z.c  (the score)
        v8f acc;
        #pragma unroll
        for (int v = 0; v < 8; ++v) acc[v] = nrm_q[mbase + v];

        // A[16x4] lane layout: m = lane%16, k = chunk*4 + (lane/16)*2 + {0,1}
        const int e = kt * 16 + nrow;
        #pragma unroll
        for (int kk = 0; kk < 8; ++kk) {
            v2f a = *(const v2f*)(&lds_cb[(e << 5) + swz(e, kk * 4 + khalf * 2)]);
            acc = __builtin_amdgcn_wmma_f32_16x16x4_f32(
                      false, a, false, breg[kk], (short)0, acc, false, false);
        }

        #pragma unroll
        for (int v = 0; v < 8; ++v) {
            float s = acc[v];
            if (s < minS) { minS = s; minI = mbase + v; }   // strict '<': first-min
        }
    }

    // Merge the two half-wave candidate sets (lane l holds row l%16).
    {
        float oS = __shfl_xor(minS, 16);
        int   oI = __shfl_xor(minI, 16);
        if (oS < minS || (oS == minS && oI < minI)) { minS = oS; minI = oI; }
    }

    // Emit indices (as float; d_out is a flat f32 buffer): lanes 0..15 own rows.
    if (lane < 16) {
        size_t off = (size_t)N_ROWS * E_DIM_ + 1 +
                     (size_t)(rowBase + lane) * NQ_ + q;
        out[off] = (float)minI;
    }

    // Gather zq from (unscaled) global codebook, write output slice,
    // accumulate local (zq - zc)^2.
    int   myIdx = __shfl(minI, nrow);
    const float* csrc = cbq + (size_t)myIdx * D_SUB_;
    const int d0 = khalf * 16;
    const float* zsrc = z   + (size_t)(rowBase + nrow) * E_DIM_ + q * D_SUB_ + d0;
    float*       odst = out + (size_t)(rowBase + nrow) * E_DIM_ + q * D_SUB_ + d0;

    float lpart = 0.f;
    #pragma unroll
    for (int d = 0; d < 16; ++d) {
        float c  = csrc[d0 + d];
        float zv = zsrc[d];
        odst[d]  = c;
        float df = c - zv;
        lpart += df * df;
    }

    // Wave-level shuffle reduction; one partial per wave (deterministic,
    // no atomics -> graph-replay safe).
    #pragma unroll
    for (int offm = 16; offm >= 1; offm >>= 1) lpart += __shfl_xor(lpart, offm);
    if (lane == 0) {
        int blk = blockIdx.y * gridDim.x + blockIdx.x;
        partial[blk * 8 + wave] = lpart;
    }
}

// ---------------------------------------------------------------------------
// Final deterministic loss reduction
// ---------------------------------------------------------------------------
__global__ __launch_bounds__(256) void pvq_loss(const float* __restrict__ partial,
                                                int n, float* __restrict__ out) {
    __shared__ float red[256];
    float s = 0.f;
    for (int i = threadIdx.x; i < n; i += 256) s += partial[i];
    red[threadIdx.x] = s;
    __syncthreads();
    for (int off = 128; off >= 1; off >>= 1) {
        if (threadIdx.x < off) red[threadIdx.x] += red[threadIdx.x + off];
        __syncthreads();
    }
    if (threadIdx.x == 0)
        out[(size_t)N_ROWS * E_DIM_] =
            1.25f * red[0] / ((float)N_ROWS * (float)E_DIM_);
}

// ---------------------------------------------------------------------------
extern "C" void kernel_launch(void* const* d_in, const int* in_sizes, int n_in,
                              void* d_out, int out_size, void* d_ws, size_t ws_size,
                              hipStream_t stream) {
    const float* z  = (const float*)d_in[0];   // (65536, 256) f32
    const float* cb = (const float*)d_in[1];   // (8, 512, 32) f32
    float* out = (float*)d_out;

    float* norms   = (float*)d_ws;                    // NQ*N_E floats
    float* partial = (float*)d_ws + NQ_ * N_E_;       // nBlocks*8 floats

    const int rowBlocks = N_ROWS / ROWS_PER_BLOCK;    // 512
    const int nPartial  = rowBlocks * NQ_ * 8;        // 32768

    pvq_norms<<<(NQ_ * N_E_ + 255) / 256, 256, 0, stream>>>(cb, norms);

    dim3 grid(rowBlocks, NQ_);
    pvq_main<<<grid, 256, 0, stream>>>(z, cb, norms, out, partial);

    pvq_loss<<<1, 256, 0, stream>>>(partial, nPartial, out);
}